// MesoNet_81149112091058
// MI455X (gfx1250) — compile-verified
//
#include <hip/hip_runtime.h>
#include <hip/hip_bf16.h>
#include <math.h>

// MesoNet GNN pipeline for MI455X (gfx1250, wave32).
// Heavy GEMMs run on V_WMMA_F32_16X16X4_F32 (exact f32 semantics).
// Edge message passing: msg[e] = sum_f h[e,f]*Y[src[e],f,:] + (X@B2)[src[e]],
// with Y = X @ W2^T precomputed as one dense WMMA GEMM per conv.
// GEMM inner loops are unguarded (exact grids, M%16==0, Nc%16==0) with stepped
// pointers (no per-iteration 64-bit muls); K%4!=0 handled by one clamped-index
// tail step (unconditional in-bounds loads + v_cndmask, uniform branch).

#define N_NODES 2048
#define N_EDGES 4096
#define N_BATCH 64
#define FDIM    32

typedef float v2f __attribute__((ext_vector_type(2)));
typedef float v8f __attribute__((ext_vector_type(8)));

enum { ACT_NONE = 0, ACT_RELU = 1, ACT_TANH = 2, ACT_SIG = 3 };

__device__ __forceinline__ float apply_act(float v, int act) {
    if      (act == ACT_RELU) v = v > 0.0f ? v : 0.0f;
    else if (act == ACT_TANH) v = tanhf(v);
    else if (act == ACT_SIG)  v = 1.0f / (1.0f + expf(-v));
    return v;
}

__device__ __forceinline__ v8f wmma4(v2f a, v2f b, v8f c) {
    return __builtin_amdgcn_wmma_f32_16x16x4_f32(false, a, false, b, (short)0, c,
                                                 false, false);
}

// ---------------------------------------------------------------------------
// GEMM, one wave = one 16x16 C tile. Requires M%16==0, Nc%16==0, exact grid.
// A frag (16x4 f32): lanes 0-15 -> K pair k..k+1, lanes 16-31 -> k+2..k+3.
// B frag (4x16 f32): same K split, N = lane%16.
// C/D: VGPR r -> row r + 8*(lane>>4), col = lane%16.
// ---------------------------------------------------------------------------
__global__ void gemm16_k(const float* __restrict__ A, int lda,
                         const float* __restrict__ B, int ldb,
                         float* __restrict__ C, int ldc,
                         int K, const float* __restrict__ bias, int act)
{
    const int lane = threadIdx.x & 31;
    const int half = lane >> 4;
    const int l    = lane & 15;
    const int row  = (blockIdx.y << 4) + l;
    const int col  = (blockIdx.x << 4) + l;

    const float* ap = A + (size_t)row * lda + 2 * half;
    const float* bp = B + (size_t)(2 * half) * ldb + col;
    const size_t bstep = (size_t)4 * ldb;

    v8f acc = {};
    const int K4 = K & ~3;
    for (int k = 0; k < K4; k += 4) {
        v2f a, b;
        a.x = ap[0];
        a.y = ap[1];
        b.x = bp[0];
        b.y = bp[ldb];
        acc = wmma4(a, b, acc);
        ap += 4;
        bp += bstep;
    }
    if (K4 < K) {                       // uniform branch: EXEC stays full
        const int ka = K4 + 2 * half;
        int k0 = ka     < K ? ka     : K - 1;
        int k1 = ka + 1 < K ? ka + 1 : K - 1;
        float a0 = A[(size_t)row * lda + k0];
        float a1 = A[(size_t)row * lda + k1];
        float b0 = B[(size_t)k0 * ldb + col];
        float b1 = B[(size_t)k1 * ldb + col];
        v2f a, b;
        a.x = ka     < K ? a0 : 0.0f;
        a.y = ka + 1 < K ? a1 : 0.0f;
        b.x = ka     < K ? b0 : 0.0f;
        b.y = ka + 1 < K ? b1 : 0.0f;
        acc = wmma4(a, b, acc);
    }

    const float bv = bias ? bias[col] : 0.0f;
    float* Cp = C + (size_t)((blockIdx.y << 4) + 8 * half) * ldc + col;
#pragma unroll
    for (int r = 0; r < 8; ++r)
        Cp[(size_t)r * ldc] = apply_act(acc[r] + bv, act);
}

// Wide GEMM: one wave = 32x64 C block. 2 A frags x 4 B frags -> 8 WMMA/k-step.
// Requires M%32==0, Nc%64==0.
__global__ void gemm32x64_k(const float* __restrict__ A, int lda,
                            const float* __restrict__ B, int ldb,
                            float* __restrict__ C, int ldc,
                            int K, const float* __restrict__ bias, int act)
{
    const int lane = threadIdx.x & 31;
    const int half = lane >> 4;
    const int l    = lane & 15;
    const int row0 = blockIdx.y << 5;
    const int col0 = (blockIdx.x << 6) + l;

    const float* ap0 = A + (size_t)(row0 + l) * lda + 2 * half;
    const float* ap1 = ap0 + (size_t)16 * lda;
    const float* bp  = B + (size_t)(2 * half) * ldb + col0;
    const size_t bstep = (size_t)4 * ldb;

    v8f c00 = {}, c01 = {}, c02 = {}, c03 = {};
    v8f c10 = {}, c11 = {}, c12 = {}, c13 = {};

    const int K4 = K & ~3;
    for (int k = 0; k < K4; k += 4) {
        v2f a0, a1, b0, b1, b2, b3;
        a0.x = ap0[0];  a0.y = ap0[1];
        a1.x = ap1[0];  a1.y = ap1[1];
        b0.x = bp[0];   b0.y = bp[ldb];
        b1.x = bp[16];  b1.y = bp[ldb + 16];
        b2.x = bp[32];  b2.y = bp[ldb + 32];
        b3.x = bp[48];  b3.y = bp[ldb + 48];
        c00 = wmma4(a0, b0, c00);
        c01 = wmma4(a0, b1, c01);
        c02 = wmma4(a0, b2, c02);
        c03 = wmma4(a0, b3, c03);
        c10 = wmma4(a1, b0, c10);
        c11 = wmma4(a1, b1, c11);
        c12 = wmma4(a1, b2, c12);
        c13 = wmma4(a1, b3, c13);
        ap0 += 4;
        ap1 += 4;
        bp  += bstep;
    }
    if (K4 < K) {                       // uniform branch
        const int ka = K4 + 2 * half;
        const int k0 = ka     < K ? ka     : K - 1;
        const int k1 = ka + 1 < K ? ka + 1 : K - 1;
        const bool m0 = ka     < K;
        const bool m1 = ka + 1 < K;
        const float* ar0 = A + (size_t)(row0 + l) * lda;
        const float* ar1 = ar0 + (size_t)16 * lda;
        const float* br0 = B + (size_t)k0 * ldb + col0;
        const float* br1 = B + (size_t)k1 * ldb + col0;
        v2f a0, a1, b0, b1, b2, b3;
        a0.x = m0 ? ar0[k0] : 0.0f;   a0.y = m1 ? ar0[k1] : 0.0f;
        a1.x = m0 ? ar1[k0] : 0.0f;   a1.y = m1 ? ar1[k1] : 0.0f;
        b0.x = m0 ? br0[0]  : 0.0f;   b0.y = m1 ? br1[0]  : 0.0f;
        b1.x = m0 ? br0[16] : 0.0f;   b1.y = m1 ? br1[16] : 0.0f;
        b2.x = m0 ? br0[32] : 0.0f;   b2.y = m1 ? br1[32] : 0.0f;
        b3.x = m0 ? br0[48] : 0.0f;   b3.y = m1 ? br1[48] : 0.0f;
        c00 = wmma4(a0, b0, c00);
        c01 = wmma4(a0, b1, c01);
        c02 = wmma4(a0, b2, c02);
        c03 = wmma4(a0, b3, c03);
        c10 = wmma4(a1, b0, c10);
        c11 = wmma4(a1, b1, c11);
        c12 = wmma4(a1, b2, c12);
        c13 = wmma4(a1, b3, c13);
    }

    v8f accs[2][4] = { { c00, c01, c02, c03 }, { c10, c11, c12, c13 } };
#pragma unroll
    for (int rt = 0; rt < 2; ++rt) {
        float* Cp = C + (size_t)(row0 + 16 * rt + 8 * half) * ldc + col0;
#pragma unroll
        for (int t = 0; t < 4; ++t) {
            const float bv = bias ? bias[col0 + 16 * t] : 0.0f;
#pragma unroll
            for (int r = 0; r < 8; ++r)
                Cp[(size_t)r * ldc + 16 * t] = apply_act(accs[rt][t][r] + bv, act);
        }
    }
}

// ---------------------------------------------------------------------------
// Small support kernels
// ---------------------------------------------------------------------------
__global__ void zero_k(float* p, int n) {
    int i = blockIdx.x * blockDim.x + threadIdx.x;
    if (i < n) p[i] = 0.0f;
}

// w2 (FDIM, din*dout) -> w2t (din, FDIM*dout):  w2t[i, f*dout+o] = w2[f, i*dout+o]
__global__ void transpose_w2_k(const float* __restrict__ w2, float* __restrict__ w2t,
                               int din, int dout, int total) {
    int idx = blockIdx.x * blockDim.x + threadIdx.x;
    if (idx >= total) return;
    int f   = idx / (din * dout);
    int rem = idx - f * (din * dout);
    int i   = rem / dout;
    int o   = rem - i * dout;
    w2t[(size_t)i * (FDIM * dout) + f * dout + o] = w2[idx];
}

// msg[e,o] = xb2[src[e],o] + sum_f hE[e,f] * Y[src[e], f*dout+o]; scatter to dst
__global__ void nnconv_msg_k(const float* __restrict__ hE,
                             const float* __restrict__ Y,
                             const float* __restrict__ xb2,
                             const int* __restrict__ src,
                             const int* __restrict__ dst,
                             float* __restrict__ sacc, float* __restrict__ cnt,
                             int dout) {
    int idx = blockIdx.x * blockDim.x + threadIdx.x;
    int e = idx / dout;
    if (e >= N_EDGES) return;
    int o  = idx - e * dout;
    int sn = src[e], dn = dst[e];
    const float* y  = Y + (size_t)sn * FDIM * dout + o;
    const float* he = hE + (size_t)e * FDIM;
    float acc = xb2[(size_t)sn * dout + o];
#pragma unroll 8
    for (int f = 0; f < FDIM; ++f) acc += he[f] * y[(size_t)f * dout];
    atomicAdd(&sacc[(size_t)dn * dout + o], acc);
    if (o == 0) atomicAdd(&cnt[dn], 1.0f);
}

// out = relu(R + sacc/max(cnt,1) + bias)
__global__ void nnconv_fin_k(const float* __restrict__ R, const float* __restrict__ sacc,
                             const float* __restrict__ cnt, const float* __restrict__ bias,
                             float* __restrict__ out, int dout) {
    int idx = blockIdx.x * blockDim.x + threadIdx.x;
    if (idx >= N_NODES * dout) return;
    int n = idx / dout, o = idx - n * dout;
    float c = cnt[n]; c = c > 1.0f ? c : 1.0f;
    float v = R[idx] + sacc[idx] / c + bias[o];
    out[idx] = v > 0.0f ? v : 0.0f;
}

// h_new = (1-t)*g1 + t*g2  -> z[:, 32:96]
__global__ void cfc_combine_k(const float* __restrict__ g1, const float* __restrict__ g2,
                              const float* __restrict__ tt, float* __restrict__ z) {
    int idx = blockIdx.x * blockDim.x + threadIdx.x;
    if (idx >= N_NODES * 64) return;
    int r = idx >> 6, c = idx & 63;
    float t = tt[idx];
    z[(size_t)r * 96 + 32 + c] = (1.0f - t) * g1[idx] + t * g2[idx];
}

__global__ void copy_block_k(const float* __restrict__ src, int lds_,
                             float* __restrict__ dst, int ldd, int rows, int cols) {
    int idx = blockIdx.x * blockDim.x + threadIdx.x;
    if (idx >= rows * cols) return;
    int r = idx / cols, c = idx - r * cols;
    dst[(size_t)r * ldd + c] = src[(size_t)r * lds_ + c];
}

__global__ void pool_scatter_k(const float* __restrict__ s, const int* __restrict__ batch,
                               float* __restrict__ psum, float* __restrict__ pcnt) {
    int idx = blockIdx.x * blockDim.x + threadIdx.x;
    if (idx >= N_NODES * 160) return;
    int n = idx / 160, o = idx - n * 160;
    int b = batch[n];
    atomicAdd(&psum[(size_t)b * 160 + o], s[idx]);
    if (o == 0) atomicAdd(&pcnt[b], 1.0f);
}

__global__ void pool_div_k(const float* __restrict__ psum, const float* __restrict__ pcnt,
                           float* __restrict__ pooled) {
    int idx = blockIdx.x * blockDim.x + threadIdx.x;
    if (idx >= N_BATCH * 160) return;
    int b = idx / 160;
    float c = pcnt[b]; c = c > 1.0f ? c : 1.0f;
    pooled[idx] = psum[idx] / c;
}

__global__ void fc2_k(const float* __restrict__ p1, const float* __restrict__ w,
                      const float* __restrict__ bptr, float* __restrict__ out) {
    int b = threadIdx.x;
    if (b >= N_BATCH) return;
    float acc = bptr[0];
#pragma unroll 8
    for (int k = 0; k < 128; ++k) acc += p1[(size_t)b * 128 + k] * w[k];
    out[b] = acc;
}

// ---------------------------------------------------------------------------
// Host-side helpers
// ---------------------------------------------------------------------------
static inline void gemm(hipStream_t st, const float* A, int lda, const float* B, int ldb,
                        float* C, int ldc, int M, int K, int Nc,
                        const float* bias, int act) {
    if ((Nc & 63) == 0 && (M & 31) == 0) {
        dim3 grid(Nc >> 6, M >> 5);
        gemm32x64_k<<<grid, 32, 0, st>>>(A, lda, B, ldb, C, ldc, K, bias, act);
    } else {
        dim3 grid(Nc >> 4, M >> 4);
        gemm16_k<<<grid, 32, 0, st>>>(A, lda, B, ldb, C, ldc, K, bias, act);
    }
}

struct ConvP { const float *w1, *b1, *w2, *b2, *root, *bias; };

static void run_nnconv(hipStream_t st, const float* Xin, int lda_in, int din, int dout,
                       const ConvP& p, const float* edge_attr,
                       const int* src, const int* dst,
                       float* hE, float* w2t, float* Y, float* xb2,
                       float* sacc, float* cnt, float* R, float* Out)
{
    // hE = relu(edge_attr @ w1 + b1)  : (E,12)@(12,32)
    gemm(st, edge_attr, 12, p.w1, FDIM, hE, FDIM, N_EDGES, 12, FDIM, p.b1, ACT_RELU);
    // w2t = reorder(w2) : (din, FDIM*dout)
    int tot = FDIM * din * dout;
    transpose_w2_k<<<(tot + 255) / 256, 256, 0, st>>>(p.w2, w2t, din, dout, tot);
    // Y = Xin @ w2t : (N, FDIM*dout) -- dominant GEMM (wide 32x64 path)
    gemm(st, Xin, lda_in, w2t, FDIM * dout, Y, FDIM * dout, N_NODES, din, FDIM * dout,
         nullptr, ACT_NONE);
    // xb2 = Xin @ reshape(b2, din, dout)
    gemm(st, Xin, lda_in, p.b2, dout, xb2, dout, N_NODES, din, dout, nullptr, ACT_NONE);
    // zero accumulators (d_ws is poisoned; must re-zero every call)
    zero_k<<<(N_NODES * dout + 255) / 256, 256, 0, st>>>(sacc, N_NODES * dout);
    zero_k<<<(N_NODES + 255) / 256, 256, 0, st>>>(cnt, N_NODES);
    // message + scatter mean-accumulate
    int tot2 = N_EDGES * dout;
    nnconv_msg_k<<<(tot2 + 255) / 256, 256, 0, st>>>(hE, Y, xb2, src, dst, sacc, cnt, dout);
    // R = Xin @ root
    gemm(st, Xin, lda_in, p.root, dout, R, dout, N_NODES, din, dout, nullptr, ACT_NONE);
    // Out = relu(R + sacc/max(cnt,1) + bias)
    nnconv_fin_k<<<(N_NODES * dout + 255) / 256, 256, 0, st>>>(R, sacc, cnt, p.bias, Out, dout);
}

extern "C" void kernel_launch(void* const* d_in, const int* in_sizes, int n_in,
                              void* d_out, int out_size, void* d_ws, size_t ws_size,
                              hipStream_t stream) {
    (void)in_sizes; (void)n_in; (void)out_size; (void)ws_size;
    // ---- inputs (setup_inputs dict order; params nested-dict flattened) ----
    const float* x         = (const float*)d_in[0];   // (2048,215)
    const int*   ei        = (const int*)  d_in[1];   // (2,4096)
    const float* edge_attr = (const float*)d_in[2];   // (4096,12)
    const int*   batch     = (const int*)  d_in[3];   // (2048,)
    int p = 4;
    ConvP a11, a21, sc1, sc2;
    ConvP* convs[4] = { &a11, &a21, &sc1, &sc2 };
    for (int c = 0; c < 4; ++c) {
        convs[c]->w1   = (const float*)d_in[p++];
        convs[c]->b1   = (const float*)d_in[p++];
        convs[c]->w2   = (const float*)d_in[p++];
        convs[c]->b2   = (const float*)d_in[p++];
        convs[c]->root = (const float*)d_in[p++];
        convs[c]->bias = (const float*)d_in[p++];
    }
    const float* x2_w  = (const float*)d_in[p++];
    const float* x2_b  = (const float*)d_in[p++];
    const float* x22_w = (const float*)d_in[p++];
    const float* x22_b = (const float*)d_in[p++];
    const float* xm3_w = (const float*)d_in[p++];
    const float* xm3_b = (const float*)d_in[p++];
    const float* inter_wv = (const float*)d_in[p++];
    const float* inter_wo = (const float*)d_in[p++];
    const float* gg_wv = (const float*)d_in[p++];
    const float* gg_wo = (const float*)d_in[p++];
    const float* ga_wv = (const float*)d_in[p++];
    const float* ga_wo = (const float*)d_in[p++];
    const float* cfc_wg = (const float*)d_in[p++];
    const float* cfc_bg = (const float*)d_in[p++];
    const float* cfc_wh = (const float*)d_in[p++];
    const float* cfc_bh = (const float*)d_in[p++];
    const float* cfc_wt = (const float*)d_in[p++];
    const float* cfc_bt = (const float*)d_in[p++];
    const float* cfc_wout = (const float*)d_in[p++];
    const float* fc1_w = (const float*)d_in[p++];
    const float* fc1_b = (const float*)d_in[p++];
    const float* fc2_w = (const float*)d_in[p++];
    const float* fc2_b = (const float*)d_in[p++];

    const int* src = ei;             // edge_index[0]
    const int* dst = ei + N_EDGES;   // edge_index[1]
    float* out = (float*)d_out;

    // ---- workspace carve (floats) ----
    float* ws = (float*)d_ws;
    size_t off = 0;
    auto take = [&](size_t n) { float* q = ws + off; off += n; return q; };
    float* Y       = take((size_t)N_NODES * FDIM * 160);  // reused by all convs
    float* w2t     = take((size_t)160 * FDIM * 160);
    float* hE      = take((size_t)N_EDGES * FDIM);
    float* sacc    = take((size_t)N_NODES * 160);
    float* cnt     = take((size_t)N_NODES);
    float* Rtmp    = take((size_t)N_NODES * 160);
    float* xb2     = take((size_t)N_NODES * 160);
    float* x1      = take((size_t)N_NODES * 32);
    float* x2f     = take((size_t)N_NODES * 32);
    float* tA      = take((size_t)N_NODES * 32);
    float* inter   = take((size_t)N_NODES * 32);
    float* glob    = take((size_t)N_NODES * 32);
    float* z       = take((size_t)N_NODES * 96);          // [u(32) | h(64)]
    float* g1      = take((size_t)N_NODES * 64);
    float* g2      = take((size_t)N_NODES * 64);
    float* tt      = take((size_t)N_NODES * 64);
    float* steps   = take((size_t)N_NODES * 96);
    float* x2o_pre = take((size_t)N_NODES * 96);
    float* x2o     = take((size_t)N_NODES * 96);
    float* xmcat   = take((size_t)N_NODES * 160);
    float* xm      = take((size_t)N_NODES * 160);
    float* s1      = take((size_t)N_NODES * 160);
    float* s2      = take((size_t)N_NODES * 160);
    float* psum    = take((size_t)N_BATCH * 160);
    float* pcnt    = take((size_t)N_BATCH);
    float* pooled  = take((size_t)N_BATCH * 160);
    float* p1      = take((size_t)N_BATCH * 128);

    // ---- stage 1: x1 = relu(nnconv_a11(x[:, :41])) ----
    run_nnconv(stream, x, 215, 41, 32, a11, edge_attr, src, dst,
               hE, w2t, Y, xb2, sacc, cnt, Rtmp, x1);

    // ---- x2f = relu(x[:,41:47] @ x2_w + x2_b) ----
    gemm(stream, x + 41, 215, x2_w, 32, x2f, 32, N_NODES, 6, 32, x2_b, ACT_RELU);

    // ---- G chain: inter -> glob -> grp ----
    gemm(stream, x + 160, 215, inter_wv, 32, tA, 32, N_NODES, 55, 32, nullptr, ACT_NONE);
    gemm(stream, tA, 32, inter_wo, 32, inter, 32, N_NODES, 32, 32, nullptr, ACT_NONE);
    gemm(stream, inter, 32, gg_wv, 32, tA, 32, N_NODES, 32, 32, nullptr, ACT_NONE);
    gemm(stream, tA, 32, gg_wo, 32, glob, 32, N_NODES, 32, 32, nullptr, ACT_NONE);
    gemm(stream, glob, 32, ga_wv, 32, tA, 32, N_NODES, 32, 32, nullptr, ACT_NONE);
    // grp straight into z[:,32:64]; glob copied into z[:,64:96]; u=x2f into z[:,0:32]
    gemm(stream, tA, 32, ga_wo, 32, z + 32, 96, N_NODES, 32, 32, nullptr, ACT_NONE);
    copy_block_k<<<(N_NODES * 32 + 255) / 256, 256, 0, stream>>>(glob, 32, z + 64, 96, N_NODES, 32);
    copy_block_k<<<(N_NODES * 32 + 255) / 256, 256, 0, stream>>>(x2f, 32, z, 96, N_NODES, 32);

    // ---- CFC recurrence (3 steps) ----
    for (int it = 0; it < 3; ++it) {
        gemm(stream, z, 96, cfc_wg, 64, g1, 64, N_NODES, 96, 64, cfc_bg, ACT_TANH);
        gemm(stream, z, 96, cfc_wh, 64, g2, 64, N_NODES, 96, 64, cfc_bh, ACT_TANH);
        gemm(stream, z, 96, cfc_wt, 64, tt, 64, N_NODES, 96, 64, cfc_bt, ACT_SIG);
        cfc_combine_k<<<(N_NODES * 64 + 255) / 256, 256, 0, stream>>>(g1, g2, tt, z);
        // u = h @ cfc_wout -> steps[:, 32*it] and back into z[:,0:32]
        gemm(stream, z + 32, 96, cfc_wout, 32, steps + 32 * it, 96, N_NODES, 64, 32,
             nullptr, ACT_NONE);
        copy_block_k<<<(N_NODES * 32 + 255) / 256, 256, 0, stream>>>(steps + 32 * it, 96, z, 96,
                                                                     N_NODES, 32);
    }

    // ---- x2o_pre = relu(steps @ x22_w + x22_b) ----
    gemm(stream, steps, 96, x22_w, 96, x2o_pre, 96, N_NODES, 96, 96, x22_b, ACT_RELU);

    // ---- x2o = relu(nnconv_a21(x2o_pre)) ----
    run_nnconv(stream, x2o_pre, 96, 96, 96, a21, edge_attr, src, dst,
               hE, w2t, Y, xb2, sacc, cnt, Rtmp, x2o);

    // ---- xm = relu([x2o | x1 | glob] @ xm3_w + xm3_b) ----
    copy_block_k<<<(N_NODES * 96 + 255) / 256, 256, 0, stream>>>(x2o, 96, xmcat, 160, N_NODES, 96);
    copy_block_k<<<(N_NODES * 32 + 255) / 256, 256, 0, stream>>>(x1, 32, xmcat + 96, 160, N_NODES, 32);
    copy_block_k<<<(N_NODES * 32 + 255) / 256, 256, 0, stream>>>(glob, 32, xmcat + 128, 160, N_NODES, 32);
    gemm(stream, xmcat, 160, xm3_w, 160, xm, 160, N_NODES, 160, 160, xm3_b, ACT_RELU);

    // ---- s = relu(nnconv_sc1(xm)); s = relu(nnconv_sc2(s)) ----
    run_nnconv(stream, xm, 160, 160, 160, sc1, edge_attr, src, dst,
               hE, w2t, Y, xb2, sacc, cnt, Rtmp, s1);
    run_nnconv(stream, s1, 160, 160, 160, sc2, edge_attr, src, dst,
               hE, w2t, Y, xb2, sacc, cnt, Rtmp, s2);

    // ---- mean pool over batch, then fc1 -> fc2 ----
    zero_k<<<(N_BATCH * 160 + 255) / 256, 256, 0, stream>>>(psum, N_BATCH * 160);
    zero_k<<<1, 64, 0, stream>>>(pcnt, N_BATCH);
    pool_scatter_k<<<(N_NODES * 160 + 255) / 256, 256, 0, stream>>>(s2, batch, psum, pcnt);
    pool_div_k<<<(N_BATCH * 160 + 255) / 256, 256, 0, stream>>>(psum, pcnt, pooled);
    gemm(stream, pooled, 160, fc1_w, 128, p1, 128, N_BATCH, 160, 128, fc1_b, ACT_RELU);
    fc2_k<<<1, 64, 0, stream>>>(p1, fc2_w, fc2_b, out);
}